// ECE_Loss_9337258901735
// MI455X (gfx1250) — compile-verified
//
#include <hip/hip_runtime.h>
#include <cstdint>

#define NUM_CLASSES     128
#define ROWS_PER_CHUNK  8
#define WAVES_PER_BLOCK 4
#define BLOCK_THREADS   (WAVES_PER_BLOCK * 32)
#define MAX_NBLK        2048
#define ROW_PAD_DW      132                      // 128 data + 4 pad dwords (bank-conflict-free)
#define ROW_PAD_BYTES   (ROW_PAD_DW * 4)         // 528
#define TILE_DW         (ROWS_PER_CHUNK * ROW_PAD_DW)   // 1056 dwords = 4224 B per buffer
#define FIN_SLICES      32

// ---------------------------------------------------------------------------
// Async copy of one 8-row (4 KB) logits tile into LDS with padded row stride.
// Each of the 8 instructions moves one 512 B row: lane l moves 16 bytes.
// Tracked by ASYNCcnt (CDNA5 async global->LDS path).
// ---------------------------------------------------------------------------
__device__ __forceinline__ void async_tile_load(uint32_t ldsBase, const float* gBase, int lane)
{
#pragma unroll
  for (int j = 0; j < ROWS_PER_CHUNK; ++j) {
    uint32_t laddr = ldsBase + (uint32_t)(j * ROW_PAD_BYTES) + (uint32_t)(lane * 16);
    uint64_t gaddr = (uint64_t)(uintptr_t)gBase + (uint64_t)(j * 512 + lane * 16);
    asm volatile("global_load_async_to_lds_b128 %0, %1, off"
                 :: "v"(laddr), "v"(gaddr)
                 : "memory");
  }
}

// ---------------------------------------------------------------------------
// bin = searchsorted(edges[1:], conf, side='left') clipped to [0,9].
// Since 0 < conf <= 1 this is exactly the count of upper edges < conf.
// Accumulate (cnt, conf_sum, acc_sum) into this wave's private LDS bins
// with hardware ds_add_f32 (layout: [0..9]=cnt, [10..19]=conf, [20..29]=acc).
// ---------------------------------------------------------------------------
__device__ __forceinline__ void bin_accumulate(uint32_t binsBase, float conf, float acc)
{
  int bin = 0;
  bin += conf > 0.1f; bin += conf > 0.2f; bin += conf > 0.3f;
  bin += conf > 0.4f; bin += conf > 0.5f; bin += conf > 0.6f;
  bin += conf > 0.7f; bin += conf > 0.8f; bin += conf > 0.9f;
  uint32_t a = binsBase + (uint32_t)(bin * 4);
  float one = 1.0f;
  asm volatile("ds_add_f32 %0, %1"           :: "v"(a), "v"(one)  : "memory");
  asm volatile("ds_add_f32 %0, %1 offset:40" :: "v"(a), "v"(conf) : "memory");
  asm volatile("ds_add_f32 %0, %1 offset:80" :: "v"(a), "v"(acc)  : "memory");
}

__global__ __launch_bounds__(BLOCK_THREADS)
void ece_main(const float* __restrict__ logits,
              const long long* __restrict__ targets,
              float* __restrict__ partials,   // [nblk][32]
              int N, int nchunks, int totalWaves)
{
  __shared__ float tiles[WAVES_PER_BLOCK][2][TILE_DW];
  __shared__ float bins[WAVES_PER_BLOCK][32];

  const int tid  = threadIdx.x;
  const int wave = tid >> 5;
  const int lane = tid & 31;
  const int g    = lane >> 2;   // row within 8-row chunk
  const int sub  = lane & 3;    // 32-float slice within row

  bins[wave][lane] = 0.0f;      // wave-private, in-order with later ds_add

  const uint32_t binsBase = (uint32_t)(uintptr_t)&bins[wave][0];
  const uint32_t tile0    = (uint32_t)(uintptr_t)&tiles[wave][0][0];
  const uint32_t tile1    = (uint32_t)(uintptr_t)&tiles[wave][1][0];

  const int waveGid = blockIdx.x * WAVES_PER_BLOCK + wave;

  int c = waveGid;
  if (c < nchunks) async_tile_load(tile0, logits + (size_t)c * 1024, lane);

  int buf = 0;
  for (; c < nchunks; c += totalWaves) {
    const int cn = c + totalWaves;
    if (cn < nchunks) {
      // prefetch next tile into the other buffer, then wait only for current tile
      async_tile_load(buf ? tile0 : tile1, logits + (size_t)cn * 1024, lane);
      asm volatile("s_wait_asynccnt 8" ::: "memory");
    } else {
      asm volatile("s_wait_asynccnt 0" ::: "memory");
    }

    // each lane reads its 32 floats (8 x b128, conflict-free via 132-dw stride)
    const float4* lrow = (const float4*)&tiles[wave][buf][g * ROW_PAD_DW + sub * 32];
    float4 v[8];
#pragma unroll
    for (int k = 0; k < 8; ++k) v[k] = lrow[k];

    // pass 1: row max (lane-local fold + 2-step butterfly across the 4-lane group)
    float m = v[0].x;
#pragma unroll
    for (int k = 0; k < 8; ++k)
      m = fmaxf(m, fmaxf(fmaxf(v[k].x, v[k].y), fmaxf(v[k].z, v[k].w)));
    m = fmaxf(m, __shfl_xor(m, 1, 32));
    m = fmaxf(m, __shfl_xor(m, 2, 32));

    // pass 2: sum of exp(l - m)  (data still in registers)
    float s = 0.0f;
#pragma unroll
    for (int k = 0; k < 8; ++k) {
      s += __expf(v[k].x - m);
      s += __expf(v[k].y - m);
      s += __expf(v[k].z - m);
      s += __expf(v[k].w - m);
    }
    s += __shfl_xor(s, 1, 32);
    s += __shfl_xor(s, 2, 32);

    if (sub == 0) {                       // one lane finalizes each row
      const float conf = 1.0f / s;        // conf = max softmax prob, s >= 1
      const int   row  = c * ROWS_PER_CHUNK + g;
      const int   t    = (int)targets[row];
      const float tv   = logits[(size_t)row * NUM_CLASSES + t];  // L2-hot
      const float acc  = (tv == m) ? 1.0f : 0.0f;  // pred==target <=> target hits row max
      bin_accumulate(binsBase, conf, acc);
    }
    buf ^= 1;
  }

  // leftover rows (N % 8), handled scalar by block 0 / wave 0
  if (blockIdx.x == 0 && wave == 0) {
    const int done = nchunks * ROWS_PER_CHUNK;
    const int rem  = N - done;
    if (lane < rem) {
      const int row = done + lane;
      const float* p = logits + (size_t)row * NUM_CLASSES;
      float m = -INFINITY;
#pragma unroll 1
      for (int k = 0; k < NUM_CLASSES; ++k) m = fmaxf(m, p[k]);
      float s = 0.0f;
#pragma unroll 1
      for (int k = 0; k < NUM_CLASSES; ++k) s += __expf(p[k] - m);
      const float conf = 1.0f / s;
      const int   t    = (int)targets[row];
      const float acc  = (p[t] == m) ? 1.0f : 0.0f;
      bin_accumulate(binsBase, conf, acc);
    }
  }

  asm volatile("s_wait_dscnt 0" ::: "memory");
  __syncthreads();

  // block-level reduce of per-wave bins -> per-block partials (plain stores, no atomics)
  if (tid < 32) {
    float s = 0.0f;
    if (tid < 30) {
#pragma unroll
      for (int w = 0; w < WAVES_PER_BLOCK; ++w) s += bins[w][tid];
    }
    partials[(size_t)blockIdx.x * 32 + tid] = s;
  }
}

// ---------------------------------------------------------------------------
// Parallel finalize: 32 slices x 32 components. Each thread sums ~nblk/32
// L2-hot partials (coalesced: a wave spans 32 consecutive components = one
// 128B line per iteration), then a 32-wide LDS fold and the 10-bin epilogue.
// ---------------------------------------------------------------------------
__global__ __launch_bounds__(FIN_SLICES * 32)
void ece_finalize(const float* __restrict__ partials,
                  float* __restrict__ out, int N, int nblk)
{
  __shared__ float red[FIN_SLICES][32];
  const int comp  = threadIdx.x & 31;
  const int slice = threadIdx.x >> 5;

  float s = 0.0f;
#pragma unroll 8
  for (int b = slice; b < nblk; b += FIN_SLICES)
    s += partials[(size_t)b * 32 + comp];
  red[slice][comp] = s;
  __syncthreads();

  if (threadIdx.x < 32) {
    float t = 0.0f;
#pragma unroll
    for (int k = 0; k < FIN_SLICES; ++k) t += red[k][threadIdx.x];
    red[0][threadIdx.x] = t;   // own-column read-then-write only; safe
  }
  __syncthreads();

  if (threadIdx.x == 0) {
    float ece = 0.0f;
#pragma unroll
    for (int i = 0; i < 10; ++i) {
      const float c  = red[0][i];
      const float cs = red[0][10 + i];
      const float as = red[0][20 + i];
      if (c > 0.0f) ece += fabsf(cs / c - as / c) * (c / (float)N);
    }
    out[0] = ece;
  }
}

extern "C" void kernel_launch(void* const* d_in, const int* in_sizes, int n_in,
                              void* d_out, int out_size, void* d_ws, size_t ws_size,
                              hipStream_t stream)
{
  const float*     logits  = (const float*)d_in[0];
  const long long* targets = (const long long*)d_in[1];
  float*           out     = (float*)d_out;
  float*           part    = (float*)d_ws;

  const int N = in_sizes[1];                 // number of rows (targets count)
  const int nchunks = N / ROWS_PER_CHUNK;

  int nblk = MAX_NBLK;
  const size_t need = (size_t)nblk * 32 * sizeof(float);
  if (ws_size < need) {
    nblk = (int)(ws_size / (32 * sizeof(float)));
    if (nblk < 1) nblk = 1;
  }
  const int totalWaves = nblk * WAVES_PER_BLOCK;

  ece_main<<<nblk, BLOCK_THREADS, 0, stream>>>(logits, targets, part,
                                               N, nchunks, totalWaves);
  ece_finalize<<<1, FIN_SLICES * 32, 0, stream>>>(part, out, N, nblk);
}